// Probabilistic_NetworkNMM_48275432407245
// MI455X (gfx1250) — compile-verified
//
#include <hip/hip_runtime.h>
#include <hip/hip_bf16.h>

// Wilson-Cowan neural mass network, N=68, 200k sequential Euler steps.
// Latency-bound sequential scan -> single persistent workgroup on one WGP.
// Mat-vec conn = Cjk @ E_prev done with V_WMMA_F32_16X16X4_F32:
//   - 5 row tiles of 16 (68 padded to 80), one tile per wave (5 waves / 160 thr)
//   - K = 68 = 17 exact chunks of 4; A chunks preloaded to registers ONCE
//   - B per chunk = E_prev[k] broadcast across all 16 columns, so every
//     column of the 16x16 D accumulator equals the conn tile.
// State (E double-buffer, I, conn) lives in LDS; 2 barriers per step.
// Sigmoids use v_exp_f32 + v_rcp_f32 (hardware trans ops) to keep the
// per-step serial critical path short.

#define NN        68
#define KC        17      // 68 / 4 K-chunks per WMMA chain
#define NTHREADS  160     // 5 wave32 waves == 5 row tiles of 16

typedef __attribute__((ext_vector_type(2))) float v2f;
typedef __attribute__((ext_vector_type(8))) float v8f;

__global__ __launch_bounds__(NTHREADS)
void nmm_persistent(const float* __restrict__ params,
                    const float* __restrict__ Cjk,
                    const float* __restrict__ y0,
                    const int*   __restrict__ nsp,
                    float*       __restrict__ out)
{
    __shared__ float Ebuf[2][80];   // E history double buffer (padded)
    __shared__ float Ibuf[80];
    __shared__ float conn[80];      // rows 68..79 are zero padding

    const int tid  = threadIdx.x;
    const int wave = tid >> 5;
    const int lane = tid & 31;
    const int half = lane >> 4;            // 0: lanes 0-15, 1: lanes 16-31
    const int steps = nsp[0];

    // Uniform scalar parameters (scalar loads)
    const float tau_e = params[0], tau_i = params[1];
    const float c1 = params[2], c2 = params[3], c3 = params[4];
    const float c4 = params[5], c5 = params[6];
    const float P  = params[7], kE = params[8], kI = params[9];
    const float inv_te = 1.0f / tau_e, inv_ti = 1.0f / tau_i;

    const float aE = 1.3f, thrE = 4.0f;
    const float aI = 2.0f, thrI = 3.7f;
    const float s0e = 1.0f / (1.0f + __expf(aE * thrE));  // sigmoid(0)
    const float s0i = 1.0f / (1.0f + __expf(aI * thrI));

    // ---- init state in LDS ----
    if (tid < NN) {
        const float e0 = y0[tid];
        Ebuf[0][tid] = e0;          // E_prev(0) = y0[:N]
        Ebuf[1][tid] = e0;          // E(0)      = y0[:N]
        Ibuf[tid]    = y0[NN + tid];
    }

    // ---- preload A tiles (16x4 f32 WMMA layout) into registers, once ----
    // Lane layout per ISA: lanes 0-15 hold K = {4kc+0, 4kc+1} in (A0, A1),
    // lanes 16-31 hold K = {4kc+2, 4kc+3}.  Row m = wave*16 + (lane&15).
    const int m = wave * 16 + (lane & 15);
    float a0[KC], a1[KC];
#pragma unroll
    for (int kc = 0; kc < KC; ++kc) {
        const int k = 4 * kc + 2 * half;
        a0[kc] = (m < NN) ? Cjk[m * NN + k]     : 0.0f;  // padded rows -> 0
        a1[kc] = (m < NN) ? Cjk[m * NN + k + 1] : 0.0f;
    }
    __syncthreads();

    // ---- 200k sequential steps ----
    for (int t = 0; t < steps; ++t) {
        const int p = t & 1;
        const float* Ep = Ebuf[p];          // E_{t-1} (one-step delayed)

        // conn = Cjk @ E_prev : 17-deep f32 WMMA accumulate chain per wave
        v8f acc = {0.f, 0.f, 0.f, 0.f, 0.f, 0.f, 0.f, 0.f};
#pragma unroll
        for (int kc = 0; kc < KC; ++kc) {
            const int k = 4 * kc + 2 * half;
            v2f a; a.x = a0[kc]; a.y = a1[kc];
            v2f b; b.x = Ep[k];  b.y = Ep[k + 1];   // broadcast over 16 cols
            acc = __builtin_amdgcn_wmma_f32_16x16x4_f32(
                      false, a, false, b, (short)0, acc, false, false);
        }
        // Every column of D equals the conn tile; lane 0 holds rows 0..7,
        // lane 16 holds rows 8..15 (C/D layout: VGPR r, lane l -> M=r+8*half).
        if ((lane & 15) == 0) {
            const int rbase = wave * 16 + half * 8;
#pragma unroll
            for (int r = 0; r < 8; ++r) conn[rbase + r] = acc[r];
        }
        __syncthreads();   // conn ready; matvec reads of Ebuf[p] complete

        // Pointwise Wilson-Cowan update, one node per thread
        if (tid < NN) {
            const float E  = Ebuf[1 - p][tid];   // E_t
            const float I  = Ibuf[tid];          // I_t
            const float cn = conn[tid];

            const float xe = c1 * E - c2 * I + c5 * cn + P;
            const float se =
                __builtin_amdgcn_rcpf(1.0f + __expf(-aE * (xe - thrE))) - s0e;
            const float xi = c3 * E - c4 * I;    // Q = 0
            const float si =
                __builtin_amdgcn_rcpf(1.0f + __expf(-aI * (xi - thrI))) - s0i;

            const float En = E + (-E + (kE - E) * se) * inv_te;  // DT = 1
            const float In = I + (-I + (kI - I) * si) * inv_ti;

            Ebuf[p][tid] = En;                   // becomes E_t+1 slot
            Ibuf[tid]    = In;
            out[(size_t)tid * (size_t)steps + t] = En - In;
        }
        __syncthreads();   // protect conn/Ebuf for next iteration
    }
}

extern "C" void kernel_launch(void* const* d_in, const int* in_sizes, int n_in,
                              void* d_out, int out_size, void* d_ws, size_t ws_size,
                              hipStream_t stream) {
    const float* params = (const float*)d_in[0];
    const float* Cjk    = (const float*)d_in[1];
    const float* y0     = (const float*)d_in[2];
    // d_in[3] = Djk (unused by the reference simulation)
    const int*   nsp    = (const int*)d_in[4];

    nmm_persistent<<<1, NTHREADS, 0, stream>>>(params, Cjk, y0, nsp,
                                               (float*)d_out);
}